// RecSysGNN_4896262717867
// MI455X (gfx1250) — compile-verified
//
#include <hip/hip_runtime.h>
#include <hip/hip_bf16.h>

typedef __attribute__((ext_vector_type(2))) float v2f;
typedef __attribute__((ext_vector_type(8))) float v8f;

#define DD 64          // feature dim (fixed by the model)
#define PR_STRIDE 160  // floats per pair-row in LDS; 160 % 64 == 32 -> halves
                       // of the wave hit disjoint bank sets (conflict-free)
#define LDS_W (32 * PR_STRIDE)  // 32 pair-rows (k=0..63 in pairs) -> 20 KB

// ---------------------------------------------------------------------------
// V_WMMA_F32_16X16X4_F32 : D[16x16] = A[16x4] * B[4x16] + C, all fp32.
// ---------------------------------------------------------------------------
__device__ __forceinline__ v8f wmma_f32_k4(v2f a, v2f b, v8f c) {
  return __builtin_amdgcn_wmma_f32_16x16x4_f32(false, a, false, b, (short)0, c,
                                               false, false);
}

__device__ __forceinline__ void atomic_add_f32(float* p, float v) {
  __hip_atomic_fetch_add(p, v, __ATOMIC_RELAXED, __HIP_MEMORY_SCOPE_AGENT);
}

// Stage W (row-major [j][k], j=out col, k=in dim) into LDS pair-packed:
//   Wp[(k>>1)*PR_STRIDE + j*2 + (k&1)] = W[j][k]
// so the B fragment (W[j][k], W[j][k+1]) is one aligned ds_load_b64.
__device__ __forceinline__ void stage_weights(const float* __restrict__ W,
                                              float* __restrict__ Wp) {
  for (int i = threadIdx.x; i < DD * DD; i += blockDim.x) {
    int j = i >> 6, k = i & 63;
    Wp[(k >> 1) * PR_STRIDE + j * 2 + (k & 1)] = W[i];
  }
}

// ------------------------------- utilities ---------------------------------
__global__ void zero_f32(float* __restrict__ p, int n) {
  int i = blockIdx.x * blockDim.x + threadIdx.x;
  if (i < n) p[i] = 0.0f;
}

__global__ void deg_acc(const int* __restrict__ to, float* __restrict__ deg, int e) {
  int i = blockIdx.x * blockDim.x + threadIdx.x;
  if (i < e) atomic_add_f32(&deg[to[i]], 1.0f);
}

__global__ void deg_to_dinv(float* __restrict__ d, int n) {
  int i = blockIdx.x * blockDim.x + threadIdx.x;
  if (i < n) { float v = d[i]; d[i] = (v > 0.0f) ? rsqrtf(v) : 0.0f; }
}

__global__ void edge_norm(const int* __restrict__ frm, const int* __restrict__ to,
                          const float* __restrict__ dinv, float* __restrict__ nrm, int e) {
  int i = blockIdx.x * blockDim.x + threadIdx.x;
  if (i < e) nrm[i] = dinv[frm[i]] * dinv[to[i]];
}

// copy emb -> d_out[0:N*D], concat block 0, and working x buffer
__global__ void init_x(const float* __restrict__ emb, float* __restrict__ out0,
                       float* __restrict__ outc, float* __restrict__ xcur,
                       int nd, int cw) {
  int i = blockIdx.x * blockDim.x + threadIdx.x;
  if (i < nd) {
    float v = emb[i];
    out0[i] = v;
    xcur[i] = v;
    int node = i >> 6, c = i & 63;
    outc[node * cw + c] = v;
  }
}

// x_next = leaky_relu(agg + lin1, 0.01); also write into concat column block
__global__ void combine(const float* __restrict__ agg, const float* __restrict__ lin1,
                        float* __restrict__ xcur, float* __restrict__ outc_blk,
                        int nd, int cw) {
  int i = blockIdx.x * blockDim.x + threadIdx.x;
  if (i < nd) {
    float v = agg[i] + lin1[i];
    float y = (v > 0.0f) ? v : 0.01f * v;
    xcur[i] = y;
    int node = i >> 6, c = i & 63;
    outc_blk[node * cw + c] = y;
  }
}

// ---------------------------------------------------------------------------
// lin1 = x @ W^T + B   (fp32 GEMM via V_WMMA_F32_16X16X4_F32)
// One wave -> 16 rows x all 64 cols (4 accumulators).
// ---------------------------------------------------------------------------
__global__ void lin1_gemm(const float* __restrict__ x, const float* __restrict__ W,
                          const float* __restrict__ B, float* __restrict__ out,
                          int nTiles) {
  __shared__ float Wp[LDS_W];
  stage_weights(W, Wp);
  __syncthreads();

  int wave = threadIdx.x >> 5;
  int lane = threadIdx.x & 31;
  int tile = blockIdx.x * (blockDim.x >> 5) + wave;
  if (tile >= nTiles) return;                 // wave-uniform: EXEC stays full

  int m = lane & 15, hi = lane >> 4;
  const v2f* xr = (const v2f*)(x + (size_t)(tile * 16 + m) * DD);
  const v2f* Wp2 = (const v2f*)Wp;

  v8f z = {};
  v8f acc[4] = {z, z, z, z};
#pragma unroll
  for (int kt = 0; kt < 16; ++kt) {
    int pr = kt * 2 + hi;                     // pair-row = (kt*4 + 2*hi) >> 1
    v2f a = xr[pr];
#pragma unroll
    for (int nt = 0; nt < 4; ++nt) {
      v2f b = Wp2[pr * (PR_STRIDE / 2) + nt * 16 + m];
      acc[nt] = wmma_f32_k4(a, b, acc[nt]);
    }
  }
#pragma unroll
  for (int nt = 0; nt < 4; ++nt) {
    int col = nt * 16 + m;
    float bias = B[col];
#pragma unroll
    for (int r = 0; r < 8; ++r) {
      int row = tile * 16 + r + 8 * hi;
      out[(size_t)row * DD + col] = acc[nt][r] + bias;
    }
  }
}

// scalar tail for rows not covered by 16-row tiles (unused when N%16==0)
__global__ void lin1_tail(const float* __restrict__ x, const float* __restrict__ W,
                          const float* __restrict__ B, float* __restrict__ out,
                          int rowStart, int nElem) {
  int i = blockIdx.x * blockDim.x + threadIdx.x;
  if (i >= nElem) return;
  int row = rowStart + (i >> 6), col = i & 63;
  float s = B[col];
  for (int k = 0; k < DD; ++k) s += x[(size_t)row * DD + k] * W[col * DD + k];
  out[(size_t)row * DD + col] = s;
}

// ---------------------------------------------------------------------------
// Edge message GEMM + scatter:
//   msg[e,:] = norm[e] * (lin1[frm[e],:] + (x[frm]⊙x[to]) @ W2^T + B2)
//   agg[to[e],:] += msg[e,:]
// One wave -> 16 edges x 64 cols.
// ---------------------------------------------------------------------------
__global__ void edge_gemm(const float* __restrict__ x, const float* __restrict__ lin1,
                          const int* __restrict__ frm, const int* __restrict__ to,
                          const float* __restrict__ nrm, const float* __restrict__ W2,
                          const float* __restrict__ B2, float* __restrict__ agg,
                          int nTiles) {
  __shared__ float Wp[LDS_W];
  stage_weights(W2, Wp);
  __syncthreads();

  int wave = threadIdx.x >> 5;
  int lane = threadIdx.x & 31;
  int tile = blockIdx.x * (blockDim.x >> 5) + wave;
  if (tile >= nTiles) return;

  int m = lane & 15, hi = lane >> 4;
  int e = tile * 16 + m;
  int fr = frm[e], tt = to[e];
  const v2f* xf = (const v2f*)(x + (size_t)fr * DD);
  const v2f* xt = (const v2f*)(x + (size_t)tt * DD);
  const v2f* Wp2 = (const v2f*)Wp;

  v8f z = {};
  v8f acc[4] = {z, z, z, z};
#pragma unroll
  for (int kt = 0; kt < 16; ++kt) {
    int pr = kt * 2 + hi;
    v2f a = xf[pr] * xt[pr];                  // xj ⊙ xi fragment
#pragma unroll
    for (int nt = 0; nt < 4; ++nt) {
      v2f b = Wp2[pr * (PR_STRIDE / 2) + nt * 16 + m];
      acc[nt] = wmma_f32_k4(a, b, acc[nt]);
    }
  }

  // epilogue: +lin1[frm] +B2, *norm, scatter-add into agg[to]
#pragma unroll
  for (int r = 0; r < 8; ++r) {
    int e2 = tile * 16 + r + 8 * hi;          // C-layout row for this half-wave
    int fr2 = frm[e2];
    int to2 = to[e2];
    float nm = nrm[e2];
#pragma unroll
    for (int nt = 0; nt < 4; ++nt) {
      int col = nt * 16 + m;
      float v = nm * (lin1[(size_t)fr2 * DD + col] + acc[nt][r] + B2[col]);
      atomic_add_f32(&agg[(size_t)to2 * DD + col], v);
    }
  }
}

// scalar tail for edges not covered by 16-edge tiles (unused when E%16==0)
__global__ void edge_tail(const float* __restrict__ x, const float* __restrict__ lin1,
                          const int* __restrict__ frm, const int* __restrict__ to,
                          const float* __restrict__ nrm, const float* __restrict__ W2,
                          const float* __restrict__ B2, float* __restrict__ agg,
                          int eStart, int nElem) {
  int i = blockIdx.x * blockDim.x + threadIdx.x;
  if (i >= nElem) return;
  int e = eStart + (i >> 6), col = i & 63;
  int fr = frm[e], tt = to[e];
  float s = B2[col];
  for (int k = 0; k < DD; ++k)
    s += x[(size_t)fr * DD + k] * x[(size_t)tt * DD + k] * W2[col * DD + k];
  float v = nrm[e] * (lin1[(size_t)fr * DD + col] + s);
  atomic_add_f32(&agg[(size_t)tt * DD + col], v);
}

// ---------------------------------------------------------------------------
extern "C" void kernel_launch(void* const* d_in, const int* in_sizes, int n_in,
                              void* d_out, int out_size, void* d_ws, size_t ws_size,
                              hipStream_t stream) {
  const int E  = in_sizes[0] / 2;
  const int ND = in_sizes[1];        // N * 64
  const int N  = ND / DD;
  const int L  = in_sizes[3] / DD;   // 3

  const int*   frm = (const int*)d_in[0];
  const int*   to  = frm + E;
  const float* emb = (const float*)d_in[1];
  const float* W1  = (const float*)d_in[2];
  const float* B1  = (const float*)d_in[3];
  const float* W2  = (const float*)d_in[4];
  const float* B2  = (const float*)d_in[5];

  // workspace layout (floats)
  float* dinv = (float*)d_ws;        // N   (deg, then deg^-0.5 in place)
  float* nrm  = dinv + N;            // E
  float* lin1 = nrm + E;             // N*64
  float* agg  = lin1 + ND;           // N*64
  float* xcur = agg + ND;            // N*64

  float* out0 = (float*)d_out;       // emb copy  [N,64]
  float* outc = out0 + ND;           // concat    [N, (L+1)*64]
  const int CW = (L + 1) * DD;

  const int TPB = 256;
  const int WPB = TPB / 32;          // waves per block

  // degrees + normalization
  zero_f32<<<(N + TPB - 1) / TPB, TPB, 0, stream>>>(dinv, N);
  deg_acc<<<(E + TPB - 1) / TPB, TPB, 0, stream>>>(to, dinv, E);
  deg_to_dinv<<<(N + TPB - 1) / TPB, TPB, 0, stream>>>(dinv, N);
  edge_norm<<<(E + TPB - 1) / TPB, TPB, 0, stream>>>(frm, to, dinv, nrm, E);

  // x0 = emb; emit emb output + concat block 0
  init_x<<<(ND + TPB - 1) / TPB, TPB, 0, stream>>>(emb, out0, outc, xcur, ND, CW);

  const int nTilesN = N / 16, tailN = N % 16;
  const int nTilesE = E / 16, tailE = E % 16;

  for (int l = 0; l < L; ++l) {
    const float* Wl1 = W1 + (size_t)l * DD * DD;
    const float* Bl1 = B1 + (size_t)l * DD;
    const float* Wl2 = W2 + (size_t)l * DD * DD;
    const float* Bl2 = B2 + (size_t)l * DD;

    zero_f32<<<(ND + TPB - 1) / TPB, TPB, 0, stream>>>(agg, ND);

    lin1_gemm<<<(nTilesN + WPB - 1) / WPB, TPB, 0, stream>>>(xcur, Wl1, Bl1, lin1,
                                                             nTilesN);
    if (tailN)
      lin1_tail<<<(tailN * DD + TPB - 1) / TPB, TPB, 0, stream>>>(
          xcur, Wl1, Bl1, lin1, nTilesN * 16, tailN * DD);

    edge_gemm<<<(nTilesE + WPB - 1) / WPB, TPB, 0, stream>>>(
        xcur, lin1, frm, to, nrm, Wl2, Bl2, agg, nTilesE);
    if (tailE)
      edge_tail<<<(tailE * DD + TPB - 1) / TPB, TPB, 0, stream>>>(
          xcur, lin1, frm, to, nrm, Wl2, Bl2, agg, nTilesE * 16, tailE * DD);

    combine<<<(ND + TPB - 1) / TPB, TPB, 0, stream>>>(agg, lin1, xcur,
                                                      outc + (l + 1) * DD, ND, CW);
  }
}